// Criterion_68341519614044
// MI455X (gfx1250) — compile-verified
//
#include <hip/hip_runtime.h>
#include <math.h>

// ---------------- constants from the reference ----------------
#define MAXG 16
__device__ __constant__ float c_IMG    = 640.0f;
__device__ __constant__ float c_RADIUS = 5.0f;
#define TOPK_K 20
__device__ __constant__ float c_W_CONF = 1.5f;   // 1.0 * 1.5
__device__ __constant__ float c_W_CLS  = 1.0f;
__device__ __constant__ float c_W_REG  = 6.0f;   // 5.0 * 1.2
__device__ __constant__ float c_ALPHA  = 0.25f;
__device__ __constant__ float c_EPS    = 1e-8f;
__device__ __constant__ float c_BIG    = 100000.0f;

typedef __attribute__((ext_vector_type(2))) float v2f;
typedef __attribute__((ext_vector_type(8))) float v8f;

__device__ __forceinline__ float sigmoidf_(float x) { return 1.0f / (1.0f + expf(-x)); }

__device__ __forceinline__ float focal_(float x, float t) {
  float p  = sigmoidf_(x);
  float ce = fmaxf(x, 0.0f) - x * t + log1pf(expf(-fabsf(x)));
  float pt = p * t + (1.0f - p) * (1.0f - t);
  float om = 1.0f - pt;                       // GAMMA = 2 -> om*om
  float at = c_ALPHA * t + (1.0f - c_ALPHA) * (1.0f - t);
  return at * ce * om * om;
}

// ---- exact f32 wave(32) sum via V_WMMA_F32_16X16X4_F32 with B = ones ----
// A is 16x4 f32: lane L<16 holds A[L][0..1], lane L>=16 holds A[L-16][2..3].
// a = {x, 0}: each lane contributes x exactly once; B = ones makes every column
// of D equal to the row sums. Summing the 8 accumulator regs gives rows 0-7
// (lanes 0-15) / 8-15 (lanes 16-31) of one column; xor-16 add completes the
// 32-lane sum in every lane. Multiplies by 1.0f -> exact f32 accumulation.
__device__ __forceinline__ float wave_sum_wmma(float x) {
  v2f a; a.x = x;    a.y = 0.0f;
  v2f b; b.x = 1.0f; b.y = 1.0f;
  v8f c = {};
  c = __builtin_amdgcn_wmma_f32_16x16x4_f32(false, a, false, b, (short)0, c, false, false);
  float s = c[0] + c[1] + c[2] + c[3] + c[4] + c[5] + c[6] + c[7];
  s += __shfl_xor(s, 16, 32);
  return s;
}

// ---------------- kernel 0: zero the match bitmask ----------------
__global__ void k0_init(unsigned* __restrict__ mmask, int n) {
  int i = blockIdx.x * blockDim.x + threadIdx.x;
  if (i < n) mmask[i] = 0u;
}

// ---------------- kernel 1: cost / iou / fg0 build ----------------
__global__ void k1_cost(const float* __restrict__ conf, const float* __restrict__ cls,
                        const float* __restrict__ box,  const float* __restrict__ tboxes,
                        const int*   __restrict__ tlabels,
                        const float* __restrict__ anchors, const float* __restrict__ strides,
                        float* __restrict__ iou_buf, float* __restrict__ cost_buf,
                        float* __restrict__ fg0_buf,
                        int B, int G, int M, int NC, int chunks) {
  int b = blockIdx.x / chunks;
  int m = (blockIdx.x % chunks) * blockDim.x + threadIdx.x;

  __shared__ float sgt[MAXG * 4];
  __shared__ int   slab[MAXG];
  if (threadIdx.x < G * 4) sgt[threadIdx.x]  = tboxes[(size_t)b * G * 4 + threadIdx.x] * c_IMG;
  if (threadIdx.x < G)     slab[threadIdx.x] = tlabels[b * G + threadIdx.x];
  __syncthreads();
  if (m >= M) return;

  size_t bm = (size_t)b * M + m;
  float sconf = sigmoidf_(conf[bm]);
  const float* crow = cls + bm * (size_t)NC;

  float bnegsum = 0.0f;
  if ((NC & 3) == 0) {                        // 16B-aligned rows -> GLOBAL_LOAD_B128
    const float4* crow4 = reinterpret_cast<const float4*>(crow);
    int n4 = NC >> 2;
    for (int c4 = 0; c4 < n4; ++c4) {
      float4 v = crow4[c4];
      bnegsum += -logf(1.0f - sqrtf(sigmoidf_(v.x) * sconf) + c_EPS);
      bnegsum += -logf(1.0f - sqrtf(sigmoidf_(v.y) * sconf) + c_EPS);
      bnegsum += -logf(1.0f - sqrtf(sigmoidf_(v.z) * sconf) + c_EPS);
      bnegsum += -logf(1.0f - sqrtf(sigmoidf_(v.w) * sconf) + c_EPS);
    }
  } else {
    for (int c = 0; c < NC; ++c) {
      float p = sqrtf(sigmoidf_(crow[c]) * sconf);
      bnegsum += -logf(1.0f - p + c_EPS);
    }
  }

  float cx = anchors[2 * m], cy = anchors[2 * m + 1];
  float r  = c_RADIUS * strides[m];
  float4 pb = reinterpret_cast<const float4*>(box)[bm];
  float px1 = pb.x, py1 = pb.y, px2 = pb.z, py2 = pb.w;
  float ap  = (px2 - px1) * (py2 - py1);

  unsigned geomask = 0u;
  bool fgany = false;
  float iouv[MAXG];
#pragma unroll
  for (int g = 0; g < MAXG; ++g) {
    if (g >= G) break;
    float gx1 = sgt[g * 4], gy1 = sgt[g * 4 + 1], gx2 = sgt[g * 4 + 2], gy2 = sgt[g * 4 + 3];
    bool ib = (cx > gx1) && (cx < gx2) && (cy > gy1) && (cy < gy2);
    float gcx = (gx1 + gx2) * 0.5f, gcy = (gy1 + gy2) * 0.5f;
    bool ic = (fabsf(cx - gcx) < r) && (fabsf(cy - gcy) < r);
    if (ib && ic) geomask |= (1u << g);
    fgany = fgany || ib || ic;
    float x1 = fmaxf(gx1, px1), y1 = fmaxf(gy1, py1);
    float x2 = fminf(gx2, px2), y2 = fminf(gy2, py2);
    float inter = fmaxf(x2 - x1, 0.0f) * fmaxf(y2 - y1, 0.0f);
    float ag = (gx2 - gx1) * (gy2 - gy1);
    iouv[g] = inter / (ag + ap - inter + c_EPS);
  }
  fg0_buf[bm] = fgany ? 1.0f : 0.0f;

#pragma unroll
  for (int g = 0; g < MAXG; ++g) {
    if (g >= G) break;
    int lab = slab[g];
    float pl = sqrtf(sigmoidf_(crow[lab]) * sconf);
    float bd = -logf(pl + c_EPS) + logf(1.0f - pl + c_EPS);   // bpos - bneg at label
    float im = fgany ? iouv[g] : 0.0f;
    float cost = bnegsum + bd - 3.0f * logf(im + c_EPS)
               + c_BIG * (((geomask >> g) & 1u) ? 0.0f : 1.0f)
               + (fgany ? 0.0f : c_BIG);
    size_t o = ((size_t)(b * G + g)) * M + m;
    cost_buf[o] = cost;
    iou_buf[o]  = iouv[g];
  }
}

// ---------------- kernel 2: per-(b,g) SimOTA selection in LDS ----------------
// Phase 1: sum of top-20 masked IoUs -> dyn_k = max(1, trunc(sum)).
// Phase 2: stable k-smallest of the cost row (== rank<dyn_k under stable
//          argsort): repeated argmin with tie -> lower index; winners set bit g
//          of matchmask[b][m] via atomicOr (order-independent, deterministic).
// Reductions: wave32 xor-butterfly (ds_bpermute), 2 barriers per iteration.
__global__ void k2_assign(const float* __restrict__ iou_buf, const float* __restrict__ cost_buf,
                          const float* __restrict__ fg0_buf, unsigned* __restrict__ mmask,
                          int B, int G, int M) {
  extern __shared__ float smem[];
  float* row = smem;                 // M floats (row cache)
  __shared__ float rv[8];
  __shared__ int   ri[8];
  __shared__ int   sdynk;

  int b = blockIdx.x / G, g = blockIdx.x % G;
  size_t base = ((size_t)(b * G + g)) * M;
  int tid = threadIdx.x, lane = tid & 31, wid = tid >> 5;

  // ---- phase 1: masked IoU top-k sum ----
  for (int m = tid; m < M; m += 256)
    row[m] = (fg0_buf[(size_t)b * M + m] != 0.0f) ? iou_buf[base + m] : 0.0f;
  __syncthreads();

  float topsum = 0.0f;
  int K = (TOPK_K < M) ? TOPK_K : M;
  for (int it = 0; it < K; ++it) {
    float bv = -1.0f; int bi = -1;
    for (int m = tid; m < M; m += 256) { float v = row[m]; if (v > bv) { bv = v; bi = m; } }
#pragma unroll
    for (int off = 16; off > 0; off >>= 1) {               // wave argmax (all lanes converge)
      float ov = __shfl_xor(bv, off, 32);
      int   oi = __shfl_xor(bi, off, 32);
      if (ov > bv || (ov == bv && oi >= 0 && (bi < 0 || oi < bi))) { bv = ov; bi = oi; }
    }
    if (lane == 0) { rv[wid] = bv; ri[wid] = bi; }
    __syncthreads();
    if (tid == 0) {
      float fv = rv[0]; int fi = ri[0];
      for (int w = 1; w < 8; ++w) {
        float ov = rv[w]; int oi = ri[w];
        if (ov > fv || (ov == fv && oi >= 0 && (fi < 0 || oi < fi))) { fv = ov; fi = oi; }
      }
      if (fi >= 0) { topsum += fv; row[fi] = -1.0f; }
    }
    __syncthreads();
  }
  if (tid == 0) { int dk = (int)topsum; sdynk = (dk < 1) ? 1 : dk; }

  // ---- phase 2: dyn_k smallest costs (stable: tie -> lower index) ----
  for (int m = tid; m < M; m += 256) row[m] = cost_buf[base + m];
  __syncthreads();
  int dynk = sdynk;
  for (int it = 0; it < dynk; ++it) {
    float bv = 3.402823466e38f; int bi = M;
    for (int m = tid; m < M; m += 256) { float v = row[m]; if (v < bv) { bv = v; bi = m; } }
#pragma unroll
    for (int off = 16; off > 0; off >>= 1) {               // wave argmin, stable
      float ov = __shfl_xor(bv, off, 32);
      int   oi = __shfl_xor(bi, off, 32);
      if (ov < bv || (ov == bv && oi < bi)) { bv = ov; bi = oi; }
    }
    if (lane == 0) { rv[wid] = bv; ri[wid] = bi; }
    __syncthreads();
    if (tid == 0) {
      float fv = rv[0]; int fi = ri[0];
      for (int w = 1; w < 8; ++w) {
        float ov = rv[w]; int oi = ri[w];
        if (ov < fv || (ov == fv && oi < fi)) { fv = ov; fi = oi; }
      }
      if (fi < M) {
        atomicOr(&mmask[(size_t)b * M + fi], 1u << g);
        row[fi] = 3.402823466e38f;
      }
    }
    __syncthreads();
  }
}

// ---------------- kernel 3: match resolution + loss partials ----------------
__global__ void k3_loss(const float* __restrict__ conf, const float* __restrict__ cls,
                        const float* __restrict__ box,  const float* __restrict__ tboxes,
                        const int*   __restrict__ tlabels,
                        const float* __restrict__ iou_buf, const float* __restrict__ cost_buf,
                        const unsigned* __restrict__ mmask, float* __restrict__ partials,
                        int B, int G, int M, int NC) {
  int idx = blockIdx.x * blockDim.x + threadIdx.x;
  int total = B * M;
  float accC = 0.0f, accK = 0.0f, accB = 0.0f, accF = 0.0f;

  if (idx < total) {
    int b = idx / M, m = idx % M;
    unsigned mk = mmask[idx];
    int cnt = __popc(mk);
    float fgf = 0.0f, ioum = 0.0f;
    int gstar = 0;
    if (cnt == 1) { gstar = __ffs((int)mk) - 1; fgf = 1.0f; }
    else if (cnt > 1) {       // multi-matched: argmin cost over ALL g (first occurrence)
      float bv = 3.402823466e38f;
      for (int g = 0; g < G; ++g) {
        float c = cost_buf[((size_t)(b * G + g)) * M + m];
        if (c < bv) { bv = c; gstar = g; }
      }
      fgf = 1.0f;
    }
    if (fgf > 0.0f) ioum = iou_buf[((size_t)(b * G + gstar)) * M + m]; // raw IoU (matching*ious)

    accC = focal_(conf[idx], fgf);    // conf focal over every anchor
    accF = fgf;

    if (fgf > 0.0f) {
      int lab = tlabels[b * G + gstar];
      const float* crow = cls + (size_t)idx * NC;
      float s = 0.0f;
      if ((NC & 3) == 0) {
        const float4* crow4 = reinterpret_cast<const float4*>(crow);
        int n4 = NC >> 2;
        for (int c4 = 0; c4 < n4; ++c4) {
          float4 v = crow4[c4];
          int c = c4 << 2;
          s += focal_(v.x, (c     == lab) ? ioum : 0.0f);
          s += focal_(v.y, (c + 1 == lab) ? ioum : 0.0f);
          s += focal_(v.z, (c + 2 == lab) ? ioum : 0.0f);
          s += focal_(v.w, (c + 3 == lab) ? ioum : 0.0f);
        }
      } else {
        for (int c = 0; c < NC; ++c) s += focal_(crow[c], (c == lab) ? ioum : 0.0f);
      }
      accK = s;
      // GIoU(box_pred, gt[gstar]*IMG)
      float4 pb = reinterpret_cast<const float4*>(box)[idx];
      float b1x1 = pb.x, b1y1 = pb.y, b1x2 = pb.z, b1y2 = pb.w;
      float4 tb = reinterpret_cast<const float4*>(tboxes)[(size_t)b * G + gstar];
      float b2x1 = tb.x * c_IMG, b2y1 = tb.y * c_IMG;
      float b2x2 = tb.z * c_IMG, b2y2 = tb.w * c_IMG;
      float x1 = fmaxf(b1x1, b2x1), y1 = fmaxf(b1y1, b2y1);
      float x2 = fminf(b1x2, b2x2), y2 = fminf(b1y2, b2y2);
      float inter = fmaxf(x2 - x1, 0.0f) * fmaxf(y2 - y1, 0.0f);
      float a1 = (b1x2 - b1x1) * (b1y2 - b1y1);
      float a2 = (b2x2 - b2x1) * (b2y2 - b2y1);
      float uni = a1 + a2 - inter;
      float iou = inter / (uni + c_EPS);
      float cx1 = fminf(b1x1, b2x1), cy1 = fminf(b1y1, b2y1);
      float cx2 = fmaxf(b1x2, b2x2), cy2 = fmaxf(b1y2, b2y2);
      float ca = (cx2 - cx1) * (cy2 - cy1);
      float gi = iou - (ca - uni) / (ca + c_EPS);
      accB = 1.0f - gi;
    }
  }

  // exact f32 wave reduction via WMMA (uniform path: EXEC all-1s here)
  float wc = wave_sum_wmma(accC);
  float wk = wave_sum_wmma(accK);
  float wb = wave_sum_wmma(accB);
  float wf = wave_sum_wmma(accF);

  __shared__ float wsum[8 * 4];
  int lane = threadIdx.x & 31, wid = threadIdx.x >> 5;
  if (lane == 0) { wsum[wid * 4 + 0] = wc; wsum[wid * 4 + 1] = wk;
                   wsum[wid * 4 + 2] = wb; wsum[wid * 4 + 3] = wf; }
  __syncthreads();
  if (threadIdx.x == 0) {
    float s0 = 0, s1 = 0, s2 = 0, s3 = 0;
    for (int w = 0; w < 8; ++w) { s0 += wsum[w * 4]; s1 += wsum[w * 4 + 1];
                                  s2 += wsum[w * 4 + 2]; s3 += wsum[w * 4 + 3]; }
    partials[blockIdx.x * 4 + 0] = s0; partials[blockIdx.x * 4 + 1] = s1;
    partials[blockIdx.x * 4 + 2] = s2; partials[blockIdx.x * 4 + 3] = s3;
  }
}

// ---------------- kernel 4: final deterministic reduction + outputs ----------------
__global__ void k4_final(const float* __restrict__ partials, int nb,
                         const float* __restrict__ aw, float* __restrict__ out) {
  float a0 = 0, a1 = 0, a2 = 0, a3 = 0;
  for (int i = threadIdx.x; i < nb; i += 256) {
    a0 += partials[i * 4 + 0]; a1 += partials[i * 4 + 1];
    a2 += partials[i * 4 + 2]; a3 += partials[i * 4 + 3];
  }
  a0 = wave_sum_wmma(a0); a1 = wave_sum_wmma(a1);
  a2 = wave_sum_wmma(a2); a3 = wave_sum_wmma(a3);

  __shared__ float wsum[8 * 4];
  int lane = threadIdx.x & 31, wid = threadIdx.x >> 5;
  if (lane == 0) { wsum[wid * 4 + 0] = a0; wsum[wid * 4 + 1] = a1;
                   wsum[wid * 4 + 2] = a2; wsum[wid * 4 + 3] = a3; }
  __syncthreads();
  if (threadIdx.x == 0) {
    float s0 = 0, s1 = 0, s2 = 0, s3 = 0;
    for (int w = 0; w < 8; ++w) { s0 += wsum[w * 4]; s1 += wsum[w * 4 + 1];
                                  s2 += wsum[w * 4 + 2]; s3 += wsum[w * 4 + 3]; }
    float numfg = fmaxf(s3, 1.0f);
    float lc = s0 / numfg, lk = s1 / numfg, lb = s2 / numfg;
    float w0 = aw[0], w1 = aw[1], w2 = aw[2];
    float mx = fmaxf(w0, fmaxf(w1, w2));
    float e0 = expf(w0 - mx), e1 = expf(w1 - mx), e2 = expf(w2 - mx);
    float den = e0 + e1 + e2;
    float losses = (e0 / den) * c_W_CONF * lc + (e1 / den) * c_W_CLS * lk
                 + (e2 / den) * c_W_REG  * lb;
    out[0] = lc; out[1] = lk; out[2] = lb; out[3] = losses;
  }
}

// ---------------- launcher ----------------
extern "C" void kernel_launch(void* const* d_in, const int* in_sizes, int n_in,
                              void* d_out, int out_size, void* d_ws, size_t ws_size,
                              hipStream_t stream) {
  const float* conf    = (const float*)d_in[0];   // [B,M,1]
  const float* cls     = (const float*)d_in[1];   // [B,M,NC]
  const float* box     = (const float*)d_in[2];   // [B,M,4]
  const float* tboxes  = (const float*)d_in[3];   // [B,G,4]
  const int*   tlabels = (const int*)  d_in[4];   // [B,G]
  const float* anchors = (const float*)d_in[5];   // [M,2]
  const float* strides = (const float*)d_in[6];   // [M]
  const float* aw      = (const float*)d_in[7];   // [3]
  float* out = (float*)d_out;

  const int M  = in_sizes[5] / 2;                 // 8400
  const int B  = in_sizes[0] / M;                 // 16
  const int G  = in_sizes[4] / B;                 // 16
  const int NC = in_sizes[1] / (B * M);           // 80

  // workspace layout (~18.4 MB)
  char* w = (char*)d_ws;
  const size_t nBGM = (size_t)B * G * M;
  float*    iou_buf  = (float*)w;    w += nBGM * sizeof(float);
  float*    cost_buf = (float*)w;    w += nBGM * sizeof(float);
  float*    fg0_buf  = (float*)w;    w += (size_t)B * M * sizeof(float);
  unsigned* mmask    = (unsigned*)w; w += (size_t)B * M * sizeof(unsigned);
  const int NB3 = (B * M + 255) / 256;
  float*    partials = (float*)w;    w += (size_t)NB3 * 4 * sizeof(float);

  const int chunks = (M + 255) / 256;

  k0_init<<<(B * M + 255) / 256, 256, 0, stream>>>(mmask, B * M);
  k1_cost<<<B * chunks, 256, 0, stream>>>(conf, cls, box, tboxes, tlabels, anchors, strides,
                                          iou_buf, cost_buf, fg0_buf, B, G, M, NC, chunks);
  const size_t shmem = (size_t)M * sizeof(float);   // row cache only
  k2_assign<<<B * G, 256, shmem, stream>>>(iou_buf, cost_buf, fg0_buf, mmask, B, G, M);
  k3_loss<<<NB3, 256, 0, stream>>>(conf, cls, box, tboxes, tlabels,
                                   iou_buf, cost_buf, mmask, partials, B, G, M, NC);
  k4_final<<<1, 256, 0, stream>>>(partials, NB3, aw, out);
}